// LDPCBeliefPropagation_89069031785006
// MI455X (gfx1250) — compile-verified
//
#include <hip/hip_runtime.h>
#include <hip/hip_bf16.h>

// ---------------------------------------------------------------------------
// LDPC "belief propagation" (faithful to the reference's beta-zeroing quirk):
// collapses to an independent per-element scalar recurrence.
//   phi(x) = -ln(tanh(x/2));  phi_sum_phi_beta == phi(+inf) == -0.0 (constant)
//   m_t  = sign(llr + m_{t-1}) * (-0.0 - phi(|llr + m_{t-1}|)),  m_0 = 0
//   out  = llr + m_{max_iters}
// Memory-bound: 117 MB traffic -> ~5 us at 23.3 TB/s HBM; both buffers fit in
// the 192 MB L2, so graph replays run at L2 bandwidth. Uses gfx1250 hardware
// v_tanh_f32 / v_log_f32 (TRANS ops) and global_prefetch_b8.
// max_iters==10 is specialized to a fully unrolled straight-line path to
// remove dynamic-trip-count unroll/remainder SALU machinery.
// All amdgcn builtins are gated on __HIP_DEVICE_COMPILE__ so the host pass
// only ever sees host-compilable fallbacks (tanhf / log2f).
// ---------------------------------------------------------------------------

#define LN2_F 0.69314718055994530942f

__device__ __forceinline__ float hw_ln(float x) {
#if defined(__HIP_DEVICE_COMPILE__) && __has_builtin(__builtin_amdgcn_logf)
    return __builtin_amdgcn_logf(x) * LN2_F;   // v_log_f32 (log2) * ln2
#else
    return log2f(x) * LN2_F;                   // host-safe; device -> v_log_f32
#endif
}

// phi(beta) = -ln(tanh(beta/2)); phi(0)=+inf, phi(+inf)=-0.0, NaN-propagating
__device__ __forceinline__ float phi_f(float beta) {
#if defined(__HIP_DEVICE_COMPILE__) && __has_builtin(__builtin_amdgcn_tanhf)
    return -hw_ln(__builtin_amdgcn_tanhf(beta * 0.5f));  // v_tanh_f32 (gfx1250)
#else
    return -hw_ln(tanhf(beta * 0.5f));                   // host-safe fallback
#endif
}

__device__ __forceinline__ float bp_step(float llr, float m) {
    float v = llr + m;
    // jnp.sign semantics: sign(0)=0 (so v==0 -> 0*(-inf) = NaN, faithful)
    float alpha = (v > 0.0f) ? 1.0f : ((v < 0.0f) ? -1.0f : 0.0f);
    float phi_b = phi_f(fabsf(v));
    return alpha * (-0.0f - phi_b);    // phi(sum phi(0)) == phi(+inf) == -0.0f
}

template <int ITERS>
__device__ __forceinline__ float bp_elem_fixed(float llr) {
    float m = 0.0f;
#pragma unroll
    for (int i = 0; i < ITERS; ++i) m = bp_step(llr, m);
    return llr + m;
}

__device__ __forceinline__ float bp_elem_dyn(float llr, int iters) {
    float m = 0.0f;
#pragma unroll 1
    for (int i = 0; i < iters; ++i) m = bp_step(llr, m);
    return llr + m;
}

__global__ __launch_bounds__(256) void LDPCBeliefPropagation_kernel(
    const float* __restrict__ llr,
    const int*   __restrict__ max_iters_p,
    float*       __restrict__ out,
    int n) {
    const int iters  = *max_iters_p;
    const int n4     = n >> 2;                      // # of float4 chunks
    const int tid    = blockIdx.x * blockDim.x + threadIdx.x;
    const int stride = gridDim.x * blockDim.x;

    const float4* __restrict__ in4  = reinterpret_cast<const float4*>(llr);
    float4*       __restrict__ out4 = reinterpret_cast<float4*>(out);

    if (iters == 10) {                              // wave-uniform specialization
        for (int i = tid; i < n4; i += stride) {
            if (i + stride < n4) {
                // gfx1250: global_prefetch_b8 for next grid-stride chunk
                __builtin_prefetch(reinterpret_cast<const char*>(in4 + i + stride), 0, 0);
            }
            float4 x = in4[i];
            float4 r;                               // 4 independent chains = ILP
            r.x = bp_elem_fixed<10>(x.x);
            r.y = bp_elem_fixed<10>(x.y);
            r.z = bp_elem_fixed<10>(x.z);
            r.w = bp_elem_fixed<10>(x.w);
            out4[i] = r;
        }
        for (int i = (n4 << 2) + tid; i < n; i += stride)
            out[i] = bp_elem_fixed<10>(llr[i]);
    } else {
        for (int i = tid; i < n4; i += stride) {
            if (i + stride < n4)
                __builtin_prefetch(reinterpret_cast<const char*>(in4 + i + stride), 0, 0);
            float4 x = in4[i];
            float4 r;
            r.x = bp_elem_dyn(x.x, iters);
            r.y = bp_elem_dyn(x.y, iters);
            r.z = bp_elem_dyn(x.z, iters);
            r.w = bp_elem_dyn(x.w, iters);
            out4[i] = r;
        }
        for (int i = (n4 << 2) + tid; i < n; i += stride)
            out[i] = bp_elem_dyn(llr[i], iters);
    }
}

extern "C" void kernel_launch(void* const* d_in, const int* in_sizes, int n_in,
                              void* d_out, int out_size, void* d_ws, size_t ws_size,
                              hipStream_t stream) {
    (void)n_in; (void)d_ws; (void)ws_size; (void)out_size;

    const float* llr       = (const float*)d_in[0];   // [B,1,7] f32, flat
    const int*   max_iters = (const int*)d_in[1];     // scalar int (==10)
    float*       out       = (float*)d_out;           // [B,1,7] f32, flat
    const int n = in_sizes[0];

    const int threads = 256;                 // 8 wave32 per block
    int n4 = n >> 2;
    int blocks = (n4 + threads - 1) / threads;
    if (blocks > 4096) blocks = 4096;        // ~1M threads, ~3.5 chunks each
    if (blocks < 1) blocks = 1;

    LDPCBeliefPropagation_kernel<<<blocks, threads, 0, stream>>>(llr, max_iters, out, n);
}